// MultiHeadAttentionQuantum_65481071395980
// MI455X (gfx1250) — compile-verified
//
#include <hip/hip_runtime.h>

#define BATCH 2
#define SEQ   2048
#define EMB   256
#define HEADS 32
#define DK    8
#define ROWS  16   // rows per block in projection kernels

typedef __attribute__((ext_vector_type(2))) float v2f;
typedef __attribute__((ext_vector_type(8))) float v8f;

// ---------------------------------------------------------------------------
// Kernel 1: fused QKV projection + quantum map.
//   q/k/v[mat][b][h][s][d]  (f32) written to workspace.
// grid = (B*S/ROWS, 3), block = 256
// ---------------------------------------------------------------------------
__global__ __launch_bounds__(256)
void qkv_project_quantum(const float* __restrict__ x,
                         const float* __restrict__ Wq,
                         const float* __restrict__ Wk,
                         const float* __restrict__ Wv,
                         const float* __restrict__ theta,
                         float* __restrict__ qkv) {
  __shared__ float xs[ROWS * EMB];
  const int tid = threadIdx.x;
  const int mat = blockIdx.y;
  const float* W = (mat == 0) ? Wq : (mat == 1) ? Wk : Wv;
  const long rowbase = (long)blockIdx.x * ROWS;

  for (int i = tid; i < ROWS * EMB; i += 256)
    xs[i] = x[rowbase * EMB + i];

  float th[DK];
#pragma unroll
  for (int d = 0; d < DK; ++d) th[d] = theta[d];
  __syncthreads();

  // 16 rows x 32 heads = 512 (row, head) units; 2 per thread.
#pragma unroll
  for (int p = 0; p < 2; ++p) {
    const int unit = tid * 2 + p;
    const int rl = unit >> 5;   // local row 0..15
    const int h  = unit & 31;   // head

    float a[DK];
#pragma unroll
    for (int d = 0; d < DK; ++d) {
      const float* wr = W + (h * DK + d) * EMB;
      float acc = 0.f;
      for (int e = 0; e < EMB; ++e)
        acc = fmaf(xs[rl * EMB + e], wr[e], acc);
      a[d] = acc;
    }

    // quantum map: c = cos(a + theta); out[0] = prod(c[1:]); out[w>=1] = cumprod(c)[w]
    float c[DK];
#pragma unroll
    for (int d = 0; d < DK; ++d) c[d] = __cosf(a[d] + th[d]);
    float z0 = 1.f;
#pragma unroll
    for (int d = 1; d < DK; ++d) z0 *= c[d];
    float outv[DK];
    outv[0] = z0;
    float run = c[0];
#pragma unroll
    for (int d = 1; d < DK; ++d) { run *= c[d]; outv[d] = run; }

    const long gr = rowbase + rl;
    const int b = (int)(gr / SEQ);
    const int s = (int)(gr % SEQ);
    float* dst = qkv + ((((long)mat * BATCH + b) * HEADS + h) * SEQ + s) * DK;
#pragma unroll
    for (int d = 0; d < DK; ++d) dst[d] = outv[d];
  }
}

// ---------------------------------------------------------------------------
// Kernel 2: streaming attention per head using V_WMMA_F32_16X16X4_F32.
//
// Scores are bounded (|q|,|k| <= 1, d=8 -> |score| <= sqrt(8)), so softmax
// with a constant shift of 0 is exact and overflow-free: no running max.
// Q is pre-scaled by (1/sqrt(d))*log2(e) so P = exp2(score') needs no
// per-tile multiplies (v_exp_f32 is a native exp2).
// V is padded with a ones-column at dim 8, so the softmax denominator
// accumulates inside the same WMMA chain as the output (acc[:,8] = sum P).
//
// grid = (SEQ/16, HEADS, BATCH), block = 32 (one wave)
// ---------------------------------------------------------------------------
__global__ __launch_bounds__(32)
void attn_quantum(const float* __restrict__ qkv, float* __restrict__ ctx) {
  __shared__ float pt[16 * 16];   // P tile staging: C-layout -> A-layout
  const int lane = threadIdx.x;
  const int half = lane >> 4;     // 0: lanes 0-15, 1: lanes 16-31
  const int ln   = lane & 15;
  const int r2   = half * 2;      // K-pair base for A/B layouts
  const int qt = blockIdx.x, h = blockIdx.y, b = blockIdx.z;

  const float* qb = qkv + (((long)0 * BATCH + b) * HEADS + h) * (long)SEQ * DK;
  const float* kb = qkv + (((long)1 * BATCH + b) * HEADS + h) * (long)SEQ * DK;
  const float* vb = qkv + (((long)2 * BATCH + b) * HEADS + h) * (long)SEQ * DK;

  // Q tile in A-layout (16x4 f32): lanes0-15 hold {K0,K1}, lanes16-31 {K2,K3}
  const int qrow = qt * 16 + ln;
  v2f a0 = *(const v2f*)(qb + qrow * DK + r2);       // dims 0..3
  v2f a1 = *(const v2f*)(qb + qrow * DK + 4 + r2);   // dims 4..7
  const float qscale = 0.51010672326f;  // (1/sqrt(8)) * log2(e)
  a0.x *= qscale; a0.y *= qscale;
  a1.x *= qscale; a1.y *= qscale;

  v8f acc = {};   // O accumulator; col 8 = running softmax denominator
  const float one_col = (ln == DK) ? 1.f : 0.f;
  const int vcol = ln & 7;        // always-valid V column (pad via cndmask)

  for (int kt = 0; kt < SEQ / 16; ++kt) {
    // K tile in B-layout (4x16): B[K][n] = k[key n][dim K]
    const int krow = kt * 16 + ln;
    const v2f b0 = *(const v2f*)(kb + krow * DK + r2);
    const v2f b1 = *(const v2f*)(kb + krow * DK + 4 + r2);

    v8f sc = {};
    sc = __builtin_amdgcn_wmma_f32_16x16x4_f32(false, a0, false, b0, (short)0, sc, false, false);
    sc = __builtin_amdgcn_wmma_f32_16x16x4_f32(false, a1, false, b1, (short)0, sc, false, false);

    // P = exp2(score'); C-layout: VGPR j / lane -> row (j + 8*half), col (ln)
#pragma unroll
    for (int j = 0; j < 8; ++j)
      pt[(j + 8 * half) * 16 + ln] = __builtin_amdgcn_exp2f(sc[j]);
    __syncthreads();

    // acc += P(16x16) @ Vpad(16x16): 4 K-chunks of 4 keys each.
    // Vpad cols: 0..7 = v dims, 8 = 1.0 (denominator), 9..15 = 0.
    const float* vrow = vb + (long)(kt * 16) * DK;
#pragma unroll
    for (int kc = 0; kc < 4; ++kc) {
      const v2f pa = *(const v2f*)(&pt[ln * 16 + kc * 4 + r2]);  // A-layout reload
      const int k0 = kc * 4 + r2;
      const float vx = vrow[k0 * DK + vcol];        // unconditional, coalesced
      const float vy = vrow[(k0 + 1) * DK + vcol];
      v2f bv;
      bv.x = (ln < DK) ? vx : one_col;              // cndmask, no EXEC branch
      bv.y = (ln < DK) ? vy : one_col;
      acc = __builtin_amdgcn_wmma_f32_16x16x4_f32(false, pa, false, bv, (short)0, acc, false, false);
    }
    __syncthreads();
  }

  // normalize by denominator (col 8 of this half) and write ctx[b][row][h*8+dim]
#pragma unroll
  for (int j = 0; j < 8; ++j) {
    const float denom = __shfl(acc[j], half * 16 + DK, 32);
    if (ln < DK) {
      const int row = qt * 16 + j + 8 * half;
      ctx[(((long)b * SEQ + row) * EMB) + h * DK + ln] = acc[j] / denom;
    }
  }
}

// ---------------------------------------------------------------------------
// Kernel 3: out = ctx @ Wc^T
// grid = B*S/ROWS, block = 256 (thread owns output column)
// ---------------------------------------------------------------------------
__global__ __launch_bounds__(256)
void out_project(const float* __restrict__ ctx, const float* __restrict__ Wc,
                 float* __restrict__ out) {
  __shared__ float xs[ROWS * EMB];
  const int tid = threadIdx.x;
  const long rowbase = (long)blockIdx.x * ROWS;
  for (int i = tid; i < ROWS * EMB; i += 256)
    xs[i] = ctx[rowbase * EMB + i];
  __syncthreads();

  const float* wr = Wc + (long)tid * EMB;
  float acc[ROWS];
#pragma unroll
  for (int r = 0; r < ROWS; ++r) acc[r] = 0.f;
  for (int e = 0; e < EMB; ++e) {
    const float w = wr[e];
#pragma unroll
    for (int r = 0; r < ROWS; ++r)
      acc[r] = fmaf(xs[r * EMB + e], w, acc[r]);
  }
#pragma unroll
  for (int r = 0; r < ROWS; ++r)
    out[(rowbase + r) * EMB + tid] = acc[r];
}

// ---------------------------------------------------------------------------
extern "C" void kernel_launch(void* const* d_in, const int* in_sizes, int n_in,
                              void* d_out, int out_size, void* d_ws, size_t ws_size,
                              hipStream_t stream) {
  (void)in_sizes; (void)n_in; (void)out_size; (void)ws_size;
  const float* x     = (const float*)d_in[0];
  const float* Wq    = (const float*)d_in[1];
  const float* Wk    = (const float*)d_in[2];
  const float* Wv    = (const float*)d_in[3];
  const float* Wc    = (const float*)d_in[4];
  const float* theta = (const float*)d_in[5];
  float* out = (float*)d_out;

  float* qkv = (float*)d_ws;                                       // 12 MB
  float* ctx = qkv + (long)3 * BATCH * HEADS * SEQ * DK;           //  4 MB

  dim3 g1(BATCH * SEQ / ROWS, 3);
  qkv_project_quantum<<<g1, 256, 0, stream>>>(x, Wq, Wk, Wv, theta, qkv);

  dim3 g2(SEQ / 16, HEADS, BATCH);
  attn_quantum<<<g2, 32, 0, stream>>>(qkv, ctx);

  out_project<<<BATCH * SEQ / ROWS, 256, 0, stream>>>(ctx, Wc, out);
}